// LocalContrastLoss_90726889161514
// MI455X (gfx1250) — compile-verified
//
#include <hip/hip_runtime.h>
#include <stdint.h>

// ---------------------------------------------------------------------------
// LocalContrastLoss for MI455X (gfx1250), compile-only tuned.
//
// out = mean(|d - box7x7(d)|),  d = pred - target   (box filter is linear)
//
// Memory-bound: ~201 MB traffic -> ~8.6 us at 23.3 TB/s. Strategy:
//   * TDM (tensor_load_to_lds, iterate mode) streams 22x22 halo tiles of
//     pred/target into per-wave LDS regions with pitch 24 (TENSORcnt waits).
//     Descriptors built fully in SALU (wave index readfirstlane'd).
//   * Vertical 7-tap sums via ds_load_2addr + v_pk_add; horizontal 7-tap done
//     exactly as v(16x22) x Band(22x16) using 6 chained V_WMMA_F32_16X16X4_F32
//     (0/1 band weights -> exact in f32, matches reference numerics).
//   * Deterministic 2-kernel reduction (no float atomics) -> bit-stable
//     across graph replays.
// ---------------------------------------------------------------------------

typedef __attribute__((ext_vector_type(2)))  float        v2f;
typedef __attribute__((ext_vector_type(8)))  float        v8f;
typedef __attribute__((ext_vector_type(4)))  unsigned int v4u;
typedef __attribute__((ext_vector_type(8)))  int          v8i;
typedef __attribute__((ext_vector_type(4)))  int          v4i;

#define IMG_W        512
#define IMG_H        512
#define TILE         16
#define HALO         22            // TILE + 6
#define PITCH        24            // LDS row pitch (floats)
#define TILES_X      32            // 512/16
#define TILES_PLANE  1024          // 32*32
#define NBLOCKS      1024
#define WAVES_PB     8
#define TILES_SLOT   12            // 98304 / (1024*8)
#define INV49        (1.0f / 49.0f)
#define INV_N        (1.0f / 25165824.0f)   // 1/(32*3*512*512)

__device__ __forceinline__ int imax(int a, int b) { return a > b ? a : b; }
__device__ __forceinline__ int imin(int a, int b) { return a < b ? a : b; }

// Issue one TDM 2D tile load: w floats/row, h rows, global row stride 512
// floats, LDS row pitch 24 floats (via iterate mode).
// 6-arg clang-23 builtin variant: (g0, g1, g2, g3, <extra group: zeros>, cpol).
__device__ __forceinline__ void tdm_load_tile(unsigned lds_byte_addr,
                                              const float* gptr,
                                              int w, int h) {
  unsigned long long ga = (unsigned long long)(uintptr_t)gptr;
  v4u g0;
  g0.x = 1u;                                         // count=1, user mode
  g0.y = lds_byte_addr;                              // LDS dest (bytes)
  g0.z = (unsigned)(ga & 0xFFFFFFFFu);               // global addr lo
  g0.w = (unsigned)((ga >> 32) & 0x01FFFFFFu)        // global addr hi
       | 0x80000000u;                                // type=2 ("image")
  v8i g1;
  g1[0] = (2 << 16) | (1 << 19);     // data_size=4B, iterate_enable=1
  g1[1] = (int)((unsigned)w << 16);  // tensor_dim0[15:0] in hi half
  g1[2] = (1 << 16);                 // tensor_dim0 hi=0 | tensor_dim1=1
  g1[3] = (int)((unsigned)w << 16);  // tile_dim0 = w
  g1[4] = 1;                         // tile_dim1 = 1, tile_dim2 = 0
  g1[5] = IMG_W;                     // tensor_dim0_stride = 512 elements
  g1[6] = 0;
  g1[7] = 0;
  v4i g2;
  g2[0] = 0;
  g2[1] = PITCH;                     // lds_addr_increment per row (elements)
  g2[2] = IMG_W;                     // global_addr_increment per row
  g2[3] = (h - 1) << 16;             // iterate_count = h-1  (h iterations)
  v4i g3 = {0, 0, 0, 0};
  v8i g4 = {0, 0, 0, 0, 0, 0, 0, 0}; // extra group (clang-23 form): zeros
  __builtin_amdgcn_tensor_load_to_lds(g0, g1, g2, g3, g4, 0);
}

// Vertical 7-tap sum of d = P - T at output row m, halo column k.
__device__ __forceinline__ float vsum7(const float* __restrict__ P,
                                       const float* __restrict__ T,
                                       int m, int k) {
  float s = 0.0f;
#pragma unroll
  for (int r = 0; r < 7; ++r)
    s += P[(m + r) * PITCH + k] - T[(m + r) * PITCH + k];
  return s;
}

__global__ __launch_bounds__(256)
void lcl_main(const float* __restrict__ pred,
              const float* __restrict__ target,
              float* __restrict__ partials) {
  __shared__ float tiles[WAVES_PB][2][HALO * PITCH];  // 33792 B
  __shared__ float red[256];

  const int tid  = threadIdx.x;
  // Wave index is uniform under wave32; force it scalar so all tile math,
  // clamps and TDM descriptor words live in SGPRs (SALU, not VALU).
  const int wave = __builtin_amdgcn_readfirstlane(tid >> 5);
  const int lane = tid & 31;
  const int half = lane >> 4;       // 0: lanes 0-15, 1: lanes 16-31
  const int n    = lane & 15;       // = M for A rows, = N for B cols / C cols

  float* P = &tiles[wave][0][0];
  float* T = &tiles[wave][1][0];
  const unsigned ldsP = (unsigned)(uintptr_t)P;   // generic LDS addr low 32
  const unsigned ldsT = (unsigned)(uintptr_t)T;   // bits = LDS byte offset

  const int slot = blockIdx.x * WAVES_PB + wave;  // scalar
  float sum = 0.0f;

  for (int i = 0; i < TILES_SLOT; ++i) {
    const int t     = slot * TILES_SLOT + i;      // contiguous: L2 reuse
    const int plane = t >> 10;
    const int tp    = t & (TILES_PLANE - 1);
    const int ty    = tp >> 5;
    const int tx    = tp & (TILES_X - 1);

    const int x0 = tx * TILE - 3, y0 = ty * TILE - 3;
    const int xs = imax(x0, 0),   ys = imax(y0, 0);
    const int xe = imin(x0 + HALO, IMG_W), ye = imin(y0 + HALO, IMG_H);
    const int w  = xe - xs,       h  = ye - ys;

    // Border tiles only: zero the halo region first (scalar branch).
    if (w < HALO || h < HALO) {
      for (int j = lane; j < HALO * PITCH; j += 32) {
        P[j] = 0.0f;
        T[j] = 0.0f;
      }
    }
    // Order DS stores / prior-iteration DS reads before TDM writes LDS.
    asm volatile("s_wait_dscnt 0x0" ::: "memory");

    const unsigned off = (unsigned)(((ys - y0) * PITCH + (xs - x0)) * 4);
    const size_t   gof = (size_t)plane * (IMG_W * IMG_H)
                       + (size_t)ys * IMG_W + (size_t)xs;
    tdm_load_tile(ldsP + off, pred + gof, w, h);
    tdm_load_tile(ldsT + off, target + gof, w, h);
    __builtin_amdgcn_s_wait_tensorcnt(0);

    // Horizontal 7-tap band matmul: acc[16x16] = v[16x22] x S[22x16],
    // S[k][j] = (j <= k <= j+6).  Exact in f32 (0/1 weights).
    v8f acc = {};
#pragma unroll
    for (int c = 0; c < 6; ++c) {
      const int k0 = 4 * c + 2 * half;
      const int k1 = k0 + 1;
      v2f a, b;
      a.x = (k0 <= HALO - 1) ? vsum7(P, T, n, k0) : 0.0f;
      a.y = (k1 <= HALO - 1) ? vsum7(P, T, n, k1) : 0.0f;
      b.x = (k0 >= n && k0 <= n + 6) ? 1.0f : 0.0f;
      b.y = (k1 >= n && k1 <= n + 6) ? 1.0f : 0.0f;
      acc = __builtin_amdgcn_wmma_f32_16x16x4_f32(
          false, a, false, b, (short)0, acc, false, false);
    }

    // acc[r] holds the 49-pixel window sum at (M = r + 8*half, N = n).
#pragma unroll
    for (int r = 0; r < 8; ++r) {
      const int M  = r + 8 * half;
      const int ci = (M + 3) * PITCH + (n + 3);
      const float dc = P[ci] - T[ci];
      sum += fabsf(dc - acc[r] * INV49);
    }
  }

  // Deterministic block reduction.
  red[tid] = sum;
  __syncthreads();
#pragma unroll
  for (int k = 128; k > 0; k >>= 1) {
    if (tid < k) red[tid] += red[tid + k];
    __syncthreads();
  }
  if (tid == 0) partials[blockIdx.x] = red[0];
}

__global__ __launch_bounds__(256)
void lcl_reduce(const float* __restrict__ partials, float* __restrict__ out) {
  __shared__ float red[256];
  float s = 0.0f;
  for (int i = threadIdx.x; i < NBLOCKS; i += 256) s += partials[i];  // fixed order
  red[threadIdx.x] = s;
  __syncthreads();
#pragma unroll
  for (int k = 128; k > 0; k >>= 1) {
    if (threadIdx.x < k) red[threadIdx.x] += red[threadIdx.x + k];
    __syncthreads();
  }
  if (threadIdx.x == 0) out[0] = red[0] * INV_N;
}

extern "C" void kernel_launch(void* const* d_in, const int* in_sizes, int n_in,
                              void* d_out, int out_size, void* d_ws, size_t ws_size,
                              hipStream_t stream) {
  const float* pred   = (const float*)d_in[0];
  const float* target = (const float*)d_in[1];
  float* out      = (float*)d_out;
  float* partials = (float*)d_ws;          // 1024 floats of scratch

  lcl_main<<<NBLOCKS, 256, 0, stream>>>(pred, target, partials);
  lcl_reduce<<<1, 256, 0, stream>>>(partials, out);
}